// MeanAggregator_36386962932386
// MI455X (gfx1250) — compile-verified
//
#include <hip/hip_runtime.h>
#include <stdint.h>

// ---------------------------------------------------------------------------
// MeanAggregator: out[b, :] = mean over {10 neighbours + self} of features[idx, :]
//   B=100000, K=10, N=1e6, D=128 (fp32)
//
// Memory-bound (~620 MB @ 23.3 TB/s => ~27 us floor). Strategy:
//   * async-gather rows straight into LDS (global_load_async_to_lds_b128 /
//     ASYNCcnt) - one wave32 fetches one 512B row per instruction.
//   * compute the neighbourhood mean as a masked matmul
//     Out[16,128] = S[16,176] @ G[176,128]  via V_WMMA_F32_16X16X4_F32,
//     S block-diagonal with 1/11.  176 = 16 nodes * 11 rows (no K padding).
// ---------------------------------------------------------------------------

typedef float v2f __attribute__((ext_vector_type(2)));
typedef float v8f __attribute__((ext_vector_type(8)));

#define NODES_PER_WG 16
#define KN 10                       // sampled neighbours per node
#define FAN 11                      // KN + self
#define DIM 128                     // feature dim
#define ROWS (NODES_PER_WG * FAN)   // 176 gathered rows per workgroup
#define THREADS 256                 // 8 wave32s

__global__ __launch_bounds__(THREADS)
void mean_agg_kernel(const long long* __restrict__ nodes,
                     const long long* __restrict__ neighbours,
                     const float* __restrict__ features,
                     float* __restrict__ out,
                     int B)
{
    extern __shared__ unsigned char smem[];
    float*     G    = (float*)smem;                                  // 176*128 f32
    long long* ridx = (long long*)(smem + (size_t)ROWS * DIM * 4);   // 176 i64

    const int tid   = threadIdx.x;
    const int wave  = tid >> 5;     // wave32: 8 waves / block
    const int lane  = tid & 31;
    const int node0 = blockIdx.x * NODES_PER_WG;

    // ---- Stage A: build the 176-row neighbourhood index list --------------
    if (tid < ROWS) {
        int g = tid / FAN;
        int s = tid - g * FAN;
        int node = node0 + g;
        if (node >= B) node = B - 1;            // clamp (result row not stored)
        long long idx = (s < KN) ? neighbours[(long long)node * KN + s]
                                 : nodes[node];
        ridx[tid] = idx;
    }
    __syncthreads();

    // ---- Stage B: async-gather rows into LDS (512 B row per wave-instr) ---
    const uint32_t gbase = (uint32_t)(uintptr_t)G;    // LDS byte offset (addr[31:0])
    #pragma unroll
    for (int i = 0; i < ROWS / 8; ++i) {              // 22 rows per wave
        int r = wave * (ROWS / 8) + i;
        long long idx = ridx[r];
        uint64_t src = (uint64_t)(uintptr_t)(features + idx * DIM)
                     + (uint32_t)(lane * 16);
        uint32_t dst = gbase + (uint32_t)(r * DIM * 4 + lane * 16);
        asm volatile("global_load_async_to_lds_b128 %0, %1, off"
                     :: "v"(dst), "v"(src) : "memory");
    }
    asm volatile("s_wait_asynccnt 0x0" ::: "memory"); // own wave's copies done
    __syncthreads();                                  // all waves' copies visible

    // ---- Stage C: Out = S @ G with V_WMMA_F32_16X16X4_F32 -----------------
    // A 16x4 layout: lane holds M=lane&15; K = {0,1} (lanes 0-15) / {2,3} (16-31)
    // B 4x16 layout: VGPR0 = rows K0 (lanes 0-15) / K2 (16-31); VGPR1 = K1 / K3
    v8f acc = {0.f, 0.f, 0.f, 0.f, 0.f, 0.f, 0.f, 0.f};
    const int   n     = lane & 15;
    const int   khalf = (lane >> 4) << 1;   // 0 or 2
    const int   dbase = wave * 16;          // this wave's 16-column D tile
    const float w     = 1.0f / (float)FAN;
    const int   mlo   = (lane & 15) * FAN;  // node m owns k in [11m, 11m+11)

    for (int kb = 0; kb < ROWS; kb += 4) {
        int k0 = kb + khalf;
        int k1 = k0 + 1;
        v2f a, b;
        a.x = (k0 >= mlo && k0 < mlo + FAN) ? w : 0.0f;
        a.y = (k1 >= mlo && k1 < mlo + FAN) ? w : 0.0f;
        b.x = G[k0 * DIM + dbase + n];
        b.y = G[k1 * DIM + dbase + n];
        acc = __builtin_amdgcn_wmma_f32_16x16x4_f32(
                  /*neg_a=*/false, a, /*neg_b=*/false, b,
                  /*c_mod=*/(short)0, acc, /*reuse_a=*/false, /*reuse_b=*/false);
    }

    // ---- Stage D: store D-matrix (VGPR j: M=j lanes 0-15, M=j+8 lanes 16-31)
    const int mhi = (lane >> 4) << 3;
    #pragma unroll
    for (int j = 0; j < 8; ++j) {
        int node = node0 + mhi + j;
        if (node < B)
            out[(long long)node * DIM + dbase + n] = acc[j];
    }
}

extern "C" void kernel_launch(void* const* d_in, const int* in_sizes, int n_in,
                              void* d_out, int out_size, void* d_ws, size_t ws_size,
                              hipStream_t stream)
{
    const long long* nodes      = (const long long*)d_in[0]; // int64 [B]
    const long long* neighbours = (const long long*)d_in[1]; // int64 [B,K]
    const float*     features   = (const float*)d_in[2];     // f32   [N,D]
    float*           out        = (float*)d_out;             // f32   [B,D]

    const int B = in_sizes[0];
    const int blocks = (B + NODES_PER_WG - 1) / NODES_PER_WG;   // 6250
    const size_t shmem = (size_t)ROWS * DIM * sizeof(float)     // gathered rows
                       + (size_t)ROWS * sizeof(long long);      // index list

    mean_agg_kernel<<<blocks, THREADS, shmem, stream>>>(
        nodes, neighbours, features, out, B);
}